// CumSumSeq_45629732553371
// MI455X (gfx1250) — compile-verified
//
#include <hip/hip_runtime.h>

// csr_row from sorted COO rows == per-bin lower_bound, computed by a single
// bandwidth-bound boundary-detection pass (134 MB @ 23.3 TB/s ~ 5.8 us).
// Data movement uses the CDNA5 Tensor Data Mover (tensor_load_to_lds) to
// stage each 16 KB tile in LDS, synchronized with s_wait_tensorcnt.

typedef __attribute__((ext_vector_type(4))) unsigned int v4u;
typedef __attribute__((ext_vector_type(8))) int         v8i;
typedef __attribute__((ext_vector_type(4))) int         v4i;

#define TILE        4096   // int32 elements per block tile (16 KB LDS)
#define BLOCK       256    // 8 waves (wave32)
#define PER_THREAD  (TILE / BLOCK)

__global__ __launch_bounds__(BLOCK)
void CumSumSeq_45629732553371_kernel(const int* __restrict__ x,
                                     int* __restrict__ out,
                                     int n, int nrow) {
    __shared__ int tile[TILE];

    const int base  = (int)blockIdx.x * TILE;
    const int count = min(TILE, n - base);
    if (count <= 0) return;

#if __has_builtin(__builtin_amdgcn_tensor_load_to_lds)
    // Wave 0 issues one TDM descriptor covering the whole tile.
    if (threadIdx.x < 32u) {
        unsigned long long gaddr = (unsigned long long)(const void*)(x + base);
        unsigned int nn = (unsigned int)count;   // <= 4096, fits tile_dim0[15:0]

        // ---- D# group 0 (128b): count=1 | lds_addr | global_addr | type=2
        v4u g0;
        g0[0] = 1u;                                        // count=1, is_restore=0
        g0[1] = 0u;                                        // lds_addr = 0 (tile[] is sole LDS object)
        g0[2] = (unsigned int)(gaddr & 0xFFFFFFFFull);     // global_addr[31:0]
        g0[3] = (unsigned int)((gaddr >> 32) & 0x1FFFFFFull) // global_addr[56:32]
              | (2u << 30);                                // type = 2 ("image")

        // ---- D# group 1 (256b): data_size=4B, 1-D tensor: dim0 = tile_dim0 = nn
        v8i g1;
        g1[0] = 0x00020000;                            // data_size=2 (4B); mask=0
        g1[1] = (int)((nn & 0xFFFFu) << 16);           // tensor_dim0[15:0]  @ bits 63:48
        g1[2] = (int)((nn >> 16) & 0xFFFFu);           // tensor_dim0[31:16] @ bits 79:64
        g1[3] = (int)((nn & 0xFFFFu) << 16);           // tile_dim0          @ bits 127:112
        g1[4] = 0;                                     // tile_dim1=0, tile_dim2=0
        g1[5] = (int)nn;                               // tensor_dim0_stride low
        g1[6] = 0;
        g1[7] = 0;

        v4i z4 = {0, 0, 0, 0};                         // groups 2/3 unused (<=2-D)
#if __clang_major__ >= 23
        v8i z8 = {0, 0, 0, 0, 0, 0, 0, 0};
        __builtin_amdgcn_tensor_load_to_lds(g0, g1, z4, z4, z8, 0);
#else
        __builtin_amdgcn_tensor_load_to_lds(g0, g1, z4, z4, 0);
#endif
        __builtin_amdgcn_s_wait_tensorcnt(0);          // TENSORcnt -> 0
    }
    __asm__ volatile("" ::: "memory");
    __syncthreads();                                   // publish LDS to all 8 waves
#else
    // Fallback: plain coalesced staging through VGPRs.
    for (int l = (int)threadIdx.x; l < count; l += BLOCK) tile[l] = x[base + l];
    __syncthreads();
#endif

    // Boundary detection: thread t owns lds[t*16 .. t*16+15] (contiguous run).
    const int l0 = (int)threadIdx.x * PER_THREAD;
    if (l0 < count) {
        const int lend = min(l0 + PER_THREAD, count);
        int prev;
        if (l0 == 0) {
            prev = (base == 0) ? -1 : x[base - 1];     // one scalar global read per block
        } else {
            prev = tile[l0 - 1];
        }
        for (int l = l0; l < lend; ++l) {
            const int cur = tile[l];
            const int gi  = base + l;
            // csr_row[r] = gi for every r with x[gi-1] < r <= x[gi]
            for (int r = prev + 1; r <= cur; ++r) out[r] = gi;
            if (gi == n - 1) {
                // tail: counts past the last value, including csr_row[nrow] = n
                for (int r = cur + 1; r <= nrow; ++r) out[r] = n;
            }
            prev = cur;
        }
    }
}

extern "C" void kernel_launch(void* const* d_in, const int* in_sizes, int n_in,
                              void* d_out, int out_size, void* d_ws, size_t ws_size,
                              hipStream_t stream) {
    // d_in[0] = nrow (scalar, unused: nrow == out_size - 1), d_in[1] = sorted x
    const int* x  = (const int*)d_in[1];
    int*       out = (int*)d_out;
    const int  n    = in_sizes[1];
    const int  nrow = out_size - 1;

    const int tiles = (n + TILE - 1) / TILE;   // 8192 blocks for 33.5M elements
    CumSumSeq_45629732553371_kernel<<<tiles, BLOCK, 0, stream>>>(x, out, n, nrow);
}